// MSIndexProposal_29703993819776
// MI455X (gfx1250) — compile-verified
//
#include <hip/hip_runtime.h>
#include <stdint.h>

// Problem dims fixed by the reference setup_inputs(): scores[2,128,480,640] fp32.
constexpr int B = 2, C = 128, H = 480, W = 640;
constexpr float EPSF = 1e-6f;
constexpr float THREE_LOG2E = 4.328085122666891f;  // 3*log2(e)

constexpr int TILE = 160;       // LCM(8,16,32,40); divides H and W
constexpr int TX = W / TILE;    // 4
constexpr int TY = H / TILE;    // 3
constexpr int FWAVES = 16;      // fused kernel: 512 threads
constexpr int BWAVES = 8;       // k=24 kernel: 256 threads

// Exp-buffer block counts: pad = k - dim%k (a FULL extra k when divisible).
constexpr int HBk(int k) { return (H + (k - H % k)) / k; }
constexpr int WBk(int k) { return (W + (k - W % k)) / k; }
constexpr size_t idxSize(int k) { return (size_t)B * 2 * C * HBk(k) * WBk(k); }
constexpr size_t softSize(int k) { return (size_t)B * C * HBk(k) * WBk(k); }

// Flat output layout: indices[k0..k4] then soft_scores[k0..k4].
constexpr size_t idxOffOf(int k) {
  const int ks[5] = {8, 16, 24, 32, 40};
  size_t o = 0;
  for (int i = 0; i < 5; ++i) { if (ks[i] == k) return o; o += idxSize(ks[i]); }
  return o;
}
constexpr size_t softOffOf(int k) {
  const int ks[5] = {8, 16, 24, 32, 40};
  size_t o = 0;
  for (int i = 0; i < 5; ++i) o += idxSize(ks[i]);
  for (int i = 0; i < 5; ++i) { if (ks[i] == k) return o; o += softSize(ks[i]); }
  return o;
}

#if __has_builtin(__builtin_amdgcn_exp2f)
#define EXP2F(x) __builtin_amdgcn_exp2f(x)
#else
#define EXP2F(x) exp2f(x)
#endif
#if __has_builtin(__builtin_amdgcn_rcpf)
#define RCPF(x) __builtin_amdgcn_rcpf(x)
#else
#define RCPF(x) (1.0f / (x))
#endif

typedef unsigned int v4u __attribute__((ext_vector_type(4)));
typedef int v4i __attribute__((ext_vector_type(4)));
typedef int v8i __attribute__((ext_vector_type(8)));

// ---- wave32 reductions via ds_swizzle_b32 (group-of-32 XOR patterns) ----
template <int IMM>
__device__ __forceinline__ float swzf(float x) {
  return __int_as_float(__builtin_amdgcn_ds_swizzle(__float_as_int(x), IMM));
}
__device__ __forceinline__ float wave_max(float v) {
  v = fmaxf(v, swzf<0x041f>(v));
  v = fmaxf(v, swzf<0x081f>(v));
  v = fmaxf(v, swzf<0x101f>(v));
  v = fmaxf(v, swzf<0x201f>(v));
  v = fmaxf(v, swzf<0x401f>(v));
  return v;
}
__device__ __forceinline__ float wave_sum(float v) {
  v += swzf<0x041f>(v);
  v += swzf<0x081f>(v);
  v += swzf<0x101f>(v);
  v += swzf<0x201f>(v);
  v += swzf<0x401f>(v);
  return v;
}

// Async global -> LDS copy (16B per lane) + wait, CDNA5 async path.
__device__ __forceinline__ void async_copy_b128(uint32_t lds_addr, const float* gptr) {
  asm volatile("global_load_async_to_lds_b128 %0, %1, off"
               :: "v"(lds_addr), "v"((unsigned long long)(uintptr_t)gptr)
               : "memory");
}
__device__ __forceinline__ void wait_async0() {
  asm volatile("s_wait_asynccnt 0x0" ::: "memory");
}

// ------------------ fused kernel: k in {8,16,32,40}, one 160x160 LDS tile ------------------
template <int K, int NW>
__device__ __forceinline__ void process_blocks(const float* __restrict__ tile,
                                               float* __restrict__ out,
                                               int wv, int lane, int b, int c,
                                               int ty, int tx) {
  constexpr int NBK = TILE / K;                 // blocks per tile edge
  constexpr int HB_ = HBk(K), WB_ = WBk(K);
  constexpr size_t IOFF = idxOffOf(K), SOFF = softOffOf(K);
  constexpr int ITERS = (K * K) / 32;           // 32 | K*K for all fused K
  const size_t planeHW = (size_t)HB_ * WB_;

  for (int blk = wv; blk < NBK * NBK; blk += NW) {
    const int bi = blk / NBK;
    const int bj = blk - bi * NBK;
    const float* bp = tile + (bi * K) * TILE + bj * K;

    // Pass 1: block max of relu'd scores.
    float mx = 0.0f;
#pragma unroll
    for (int it = 0; it < ITERS; ++it) {
      const int t = lane + it * 32;
      const int i = t / K, j = t - (t / K) * K;
      mx = fmaxf(mx, fmaxf(bp[i * TILE + j], 0.0f));
    }
    mx = wave_max(mx);
    mx = fmaxf(mx, EPSF);
    const float cexp = THREE_LOG2E * RCPF(mx);

    // Pass 2: exp-weighted sums.
    float se = 0.0f, sr = 0.0f, sc = 0.0f, ss = 0.0f;
#pragma unroll
    for (int it = 0; it < ITERS; ++it) {
      const int t = lane + it * 32;
      const int i = t / K, j = t - (t / K) * K;
      const float v = fmaxf(bp[i * TILE + j], 0.0f);
      const float e = EXP2F(v * cexp);
      se += e; sr += e * (float)i; sc += e * (float)j; ss += e * v;
    }
    se = wave_sum(se);
    sr = wave_sum(sr);
    sc = wave_sum(sc);
    ss = wave_sum(ss);

    if (lane == 0) {
      const int hb = ty * NBK + bi;
      const int wb = tx * NBK + bj;
      const float inv = RCPF(se);  // se >= K*K >= 1, EPS clamp is a no-op
      float* rowPtr = out + IOFF + ((size_t)(b * 2 * C + 2 * c) * HB_ + hb) * WB_ + wb;
      rowPtr[0] = sr * inv + (float)(hb * K);
      rowPtr[planeHW] = sc * inv + (float)(wb * K);
      out[SOFF + ((size_t)(b * C + c) * HB_ + hb) * WB_ + wb] = ss * inv;
    }
  }
}

__global__ __launch_bounds__(FWAVES * 32)
void ms_fused_kernel(const float* __restrict__ scores, float* __restrict__ out) {
  __shared__ float tile[TILE * TILE];  // 100 KB of the 320 KB WGP LDS

  const int lane = threadIdx.x & 31;
  const int wv = threadIdx.x >> 5;
  int id = blockIdx.x;
  const int tx = id & (TX - 1);
  id >>= 2;
  const int ty = id % TY;
  id /= TY;
  const int c = id % C;
  const int b = id / C;

  if (wv == 0) {  // TDM ignores EXEC: exactly one wave may reach the issue point
    const unsigned long long ga = (unsigned long long)(uintptr_t)
        (scores + (((size_t)b * C + c) * H + (size_t)ty * TILE) * W + (size_t)tx * TILE);
    const uint32_t la = (uint32_t)(uintptr_t)&tile[0];  // low 32 bits == LDS offset

    // D# group 0: count=1 | lds_addr | global_addr[56:0] | type=2 ("image")
    v4u g0;
    g0[0] = 1u;
    g0[1] = la;
    g0[2] = (uint32_t)ga;
    g0[3] = (uint32_t)((ga >> 32) & 0x01FFFFFFu) | (2u << 30);
    // D# group 1: data_size=4B; tensor 640x480 stride 640; tile 160x160; no pad/multicast
    v8i g1;
    g1[0] = (int)(2u << 16);                                   // data_size code 2 = 4B
    g1[1] = (int)(((uint32_t)W & 0xFFFFu) << 16);              // tensor_dim0[15:0]
    g1[2] = (int)((((uint32_t)W >> 16) & 0xFFFFu) | (((uint32_t)H & 0xFFFFu) << 16));
    g1[3] = (int)((((uint32_t)H >> 16) & 0xFFFFu) | ((uint32_t)TILE << 16));  // tile_dim0
    g1[4] = (int)(uint32_t)TILE;                               // tile_dim1, tile_dim2=0
    g1[5] = (int)(uint32_t)W;                                  // tensor_dim0_stride[31:0]
    g1[6] = 0;
    g1[7] = 0;
    v4i zz4 = {0, 0, 0, 0};
    v8i zz8 = {0, 0, 0, 0, 0, 0, 0, 0};
#if __has_builtin(__builtin_amdgcn_tensor_load_to_lds)
    // amdgpu-toolchain (clang-23) 6-arg form: (g0 v4u, g1 v8i, v4i, v4i, v8i, cpol)
    __builtin_amdgcn_tensor_load_to_lds(g0, g1, zz4, zz4, zz8, 0);
#else
    asm volatile("tensor_load_to_lds %0, %1" :: "s"(g0), "s"(g1) : "memory");
#endif
#if __has_builtin(__builtin_amdgcn_s_wait_tensorcnt)
    __builtin_amdgcn_s_wait_tensorcnt(0);
#else
    asm volatile("s_wait_tensorcnt 0x0" ::: "memory");
#endif
    asm volatile("" ::: "memory");  // TDM wrote LDS behind the compiler's back
  }
  __syncthreads();

  process_blocks<8, FWAVES>(tile, out, wv, lane, b, c, ty, tx);
  process_blocks<16, FWAVES>(tile, out, wv, lane, b, c, ty, tx);
  process_blocks<32, FWAVES>(tile, out, wv, lane, b, c, ty, tx);
  process_blocks<40, FWAVES>(tile, out, wv, lane, b, c, ty, tx);
}

// ------------------ k=24: one wave32 per block, async b128 copy to LDS ------------------
template <int K>
__global__ __launch_bounds__(BWAVES * 32)
void ms_block_kernel(const float* __restrict__ scores, float* __restrict__ out) {
  constexpr int HB_ = HBk(K), WB_ = WBk(K);
  constexpr size_t IOFF = idxOffOf(K), SOFF = softOffOf(K);
  __shared__ float tile[BWAVES * K * K];

  const int lane = threadIdx.x & 31;
  const int wv = threadIdx.x >> 5;
  const uint32_t waveId = blockIdx.x * BWAVES + wv;
  constexpr uint32_t TOTAL = (uint32_t)B * C * HB_ * WB_;
  if (waveId >= TOTAL) return;  // wave-uniform exit

  const int wb = (int)(waveId % WB_);
  uint32_t r = waveId / WB_;
  const int hb = (int)(r % HB_);
  r /= HB_;
  const int c = (int)(r % C);
  const int b = (int)(r / C);

  const int h0 = hb * K, w0 = wb * K;
  const int hN = (H - h0) < K ? (H - h0) : K;  // K | H: full or <=0
  const int wN = (W - w0) < K ? (W - w0) : K;  // k=24: partial last col (wN=16)

  const size_t planeHW = (size_t)HB_ * WB_;
  float* rowPtr = out + IOFF + ((size_t)(b * 2 * C + 2 * c) * HB_ + hb) * WB_ + wb;
  float* colPtr = rowPtr + planeHW;
  float* softPtr = out + SOFF + ((size_t)(b * C + c) * HB_ + hb) * WB_ + wb;

  if (hN <= 0 || wN <= 0) {  // all-padding block
    if (lane == 0) { *rowPtr = (float)h0; *colPtr = (float)w0; *softPtr = 0.0f; }
    return;
  }

  const float* src = scores + (((size_t)b * C + c) * H + h0) * W + w0;
  float* myTile = &tile[wv * K * K];
  const uint32_t ldsBase = (uint32_t)(uintptr_t)myTile;

  const int n = hN * wN;
  const int n4 = n >> 2;  // wN in {24,16}: always multiple of 4 -> 16B aligned
  if (wN == K) {
    for (int t4 = lane; t4 < n4; t4 += 32) {
      const int e0 = t4 << 2;
      const int i = e0 / K, j = e0 - i * K;
      async_copy_b128(ldsBase + ((uint32_t)t4 << 4), src + (size_t)i * W + j);
    }
  } else {
    for (int t4 = lane; t4 < n4; t4 += 32) {
      const int e0 = t4 << 2;
      const int i = e0 / wN, j = e0 - i * wN;
      async_copy_b128(ldsBase + ((uint32_t)t4 << 4), src + (size_t)i * W + j);
    }
  }
  wait_async0();  // tile region is wave-private: no barrier needed

  float mx = 0.0f;
  for (int t = lane; t < n; t += 32) mx = fmaxf(mx, fmaxf(myTile[t], 0.0f));
  mx = wave_max(mx);
  mx = fmaxf(mx, EPSF);
  const float cexp = THREE_LOG2E * RCPF(mx);

  float se = 0.0f, sr = 0.0f, sc = 0.0f, ss = 0.0f;
  if (wN == K) {
    for (int t = lane; t < n; t += 32) {
      const float v = fmaxf(myTile[t], 0.0f);
      const float e = EXP2F(v * cexp);
      const int i = t / K, j = t - i * K;
      se += e; sr += e * (float)i; sc += e * (float)j; ss += e * v;
    }
  } else {
    for (int t = lane; t < n; t += 32) {
      const float v = fmaxf(myTile[t], 0.0f);
      const float e = EXP2F(v * cexp);
      const int i = t / wN, j = t - i * wN;
      se += e; sr += e * (float)i; sc += e * (float)j; ss += e * v;
    }
  }
  se = wave_sum(se);
  sr = wave_sum(sr);
  sc = wave_sum(sc);
  ss = wave_sum(ss);

  if (lane == 0) {
    const float inv = RCPF(se);
    *rowPtr = sr * inv + (float)h0;
    *colPtr = sc * inv + (float)w0;
    *softPtr = ss * inv;
  }
}

// ------------------ trivial padded blocks for the fused k's (no reads) ------------------
template <int K>
__global__ __launch_bounds__(256)
void ms_edge_kernel(float* __restrict__ out) {
  constexpr int HB_ = HBk(K), WB_ = WBk(K);
  constexpr int Hc = H / K, Wc = W / K;  // K divides H and W for fused k's
  constexpr int CNT = WB_ + Hc;          // bottom block-row (incl corner) + right block-col
  constexpr size_t IOFF = idxOffOf(K), SOFF = softOffOf(K);
  const int tid = blockIdx.x * 256 + threadIdx.x;
  if (tid >= B * C * CNT) return;
  const int t = tid % CNT;
  const int bc = tid / CNT;
  const int c = bc % C;
  const int b = bc / C;
  int hb, wb;
  if (t < WB_) { hb = Hc; wb = t; } else { hb = t - WB_; wb = Wc; }
  const size_t planeHW = (size_t)HB_ * WB_;
  float* rowPtr = out + IOFF + ((size_t)(b * 2 * C + 2 * c) * HB_ + hb) * WB_ + wb;
  rowPtr[0] = (float)(hb * K);
  rowPtr[planeHW] = (float)(wb * K);
  out[SOFF + ((size_t)(b * C + c) * HB_ + hb) * WB_ + wb] = 0.0f;
}

extern "C" void kernel_launch(void* const* d_in, const int* in_sizes, int n_in,
                              void* d_out, int out_size, void* d_ws, size_t ws_size,
                              hipStream_t stream) {
  (void)in_sizes; (void)n_in; (void)out_size; (void)d_ws; (void)ws_size;
  const float* scores = (const float*)d_in[0];
  float* out = (float*)d_out;

  // Fused pass: k in {8,16,32,40}, one TDM tile load per 160x160 region.
  hipLaunchKernelGGL(ms_fused_kernel, dim3(B * C * TY * TX), dim3(FWAVES * 32), 0,
                     stream, scores, out);

  // k=24 pass: wave-per-block with async global->LDS staging.
  {
    constexpr uint32_t TOTAL24 = (uint32_t)B * C * HBk(24) * WBk(24);
    const int grid = (int)((TOTAL24 + BWAVES - 1) / BWAVES);
    hipLaunchKernelGGL((ms_block_kernel<24>), dim3(grid), dim3(BWAVES * 32), 0,
                       stream, scores, out);
  }

  // Edge fill for the fused k's padded block-row/col (pure writes).
#define EDGE_K(KK)                                                              \
  do {                                                                          \
    constexpr int CNT_ = B * C * (WBk(KK) + H / KK);                            \
    hipLaunchKernelGGL((ms_edge_kernel<KK>), dim3((CNT_ + 255) / 256),          \
                       dim3(256), 0, stream, out);                              \
  } while (0)
  EDGE_K(8);
  EDGE_K(16);
  EDGE_K(32);
  EDGE_K(40);
#undef EDGE_K
}